// ArcModel3Phase_30442728194557
// MI455X (gfx1250) — compile-verified
//
#include <hip/hip_runtime.h>
#include <hip/hip_bf16.h>
#include <math.h>

#define NMC 1024
#define WPB 8   // waves per block (256 threads, wave32)

typedef float v2f __attribute__((ext_vector_type(2)));
typedef float v8f __attribute__((ext_vector_type(8)));

// ---------- constants ----------
#define ERFW        0.99730020f   // erf(3/sqrt(2))
#define HALF_L2PI   0.91893853f   // 0.5*log(2*pi)
#define HALF_L2OPI -0.22579135f   // 0.5*log(2/pi)
#define LOG_NMC     6.93147181f   // log(1024)
#define SQRT_2PI    2.50662827f
#define LOG2F_C     0.69314718f
#define NEG_LGAM15  0.12078224f   // -log(Gamma(1.5))
#define KSLACK      8.0f          // shift-bound slack (subsample looseness)
#define ECLAMP      80.0f         // exp-arg clamp: hard overflow guard

__device__ __forceinline__ float erfinv_f(float u) {
    float w = -__logf(fmaxf(1.0f - u * u, 1e-37f));
    float p;
    if (w < 5.0f) {
        w -= 2.5f;
        p = 2.81022636e-08f;
        p = fmaf(p, w, 3.43273939e-07f);
        p = fmaf(p, w, -3.5233877e-06f);
        p = fmaf(p, w, -4.39150654e-06f);
        p = fmaf(p, w, 0.00021858087f);
        p = fmaf(p, w, -0.00125372503f);
        p = fmaf(p, w, -0.00417768164f);
        p = fmaf(p, w, 0.246640727f);
        p = fmaf(p, w, 1.50140941f);
    } else {
        w = sqrtf(w) - 3.0f;
        p = -0.000200214257f;
        p = fmaf(p, w, 0.000100950558f);
        p = fmaf(p, w, 0.00134934322f);
        p = fmaf(p, w, -0.00367342844f);
        p = fmaf(p, w, 0.00573950773f);
        p = fmaf(p, w, -0.0076224613f);
        p = fmaf(p, w, 0.00943887047f);
        p = fmaf(p, w, 1.00167406f);
        p = fmaf(p, w, 2.83297682f);
    }
    return p * u;
}

// ---------------- kernel 1: per-(interface, MC-sample) precompute ----------------
// exp(lp[n,m]) = exp(a_n + b_m + x_m*s_n + y_m*g_n) * (1 - exp(-2*g_n*y_m))
// table entry: (s_n, g_n, a_n, -2*g_n)
__global__ void arc_precompute(const float* __restrict__ ku12,
                               const float* __restrict__ ku23,
                               const float* __restrict__ ku13,
                               const float* __restrict__ pSb,
                               const float* __restrict__ pSn,
                               const float* __restrict__ pI1,
                               const float* __restrict__ pI2,
                               const float* __restrict__ pI3,
                               float4* __restrict__ tab4) {
    int i = blockIdx.x * blockDim.x + threadIdx.x;
    if (i >= 3 * NMC) return;
    int j = i >> 10, n = i & (NMC - 1);

    float sb = pSb[0], sn = pSn[0];
    float I1 = pI1[0], I2 = pI2[0], I3 = pI3[0];
    float Ia = (j == 0) ? I1 : (j == 1) ? I2 : I1;
    float Ib = (j == 0) ? I2 : (j == 1) ? I3 : I3;
    const float* ku = (j == 0) ? ku12 : (j == 1) ? ku23 : ku13;

    float dI    = Ib - Ia;
    float Idiff = dI * ERFW;
    float Imin  = Ia + 0.5f * dI * (1.0f - ERFW);
    float tx    = fmaf(ku[n], Idiff, Imin);
    float u     = 2.0f * (tx - Ia) / dI - 1.0f;
    float e     = erfinv_f(u);
    float G     = dI / (SQRT_2PI * sb) * __expf(-e * e);

    float inv_s2 = 1.0f / (sn * sn);
    float k2     = 2.0f * inv_s2;
    float s      = tx * inv_s2;
    float g      = k2 * G;

    // all (n,m)-independent constants folded here (wf = 3.0)
    float Cj = -3.0f * __logf(sn) + HALF_L2OPI - 0.5f * __logf(k2) - __logf(6.0f * dI);
    float Dj = __logf(Idiff) - LOG_NMC + Cj;
    float a  = -0.5f * tx * tx * inv_s2 - __logf(G) - G * G * inv_s2 + e * e + Dj;

    tab4[i] = make_float4(s, g, a, -2.0f * g);
}

// ---------------- kernel 2: WMMA linear part + shifted exp-sum ----------------
__global__ __launch_bounds__(256) void arc_main(const float* __restrict__ x,
                                                const float* __restrict__ y,
                                                const float4* __restrict__ tab4,
                                                const float* __restrict__ pSn,
                                                const float* __restrict__ pI1,
                                                const float* __restrict__ pI2,
                                                const float* __restrict__ pI3,
                                                const float* __restrict__ w,
                                                float* __restrict__ partial,
                                                int Mtot) {
    __shared__ float4 lT[3 * NMC];          // 48 KB per-sample table
    __shared__ float  ldsK[WPB][16];        // per-tile shift K'
    __shared__ float  lLse[WPB][48];        // per-tile per-interface lse
    __shared__ float  red[256];

    int tid = threadIdx.x;
    for (int i = tid; i < 3 * NMC; i += 256) lT[i] = tab4[i];
    __syncthreads();

    int lane = tid & 31, wave = tid >> 5;
    int lo = lane & 15, hi = lane >> 4;
    int mbase = (blockIdx.x * WPB + wave) * 16;

    // A operands (16x4 f32 layout): lanes 0-15 hold K0/K1 for M=lane,
    // lanes 16-31 hold K2/K3 for M=lane-16.
    int mA = min(mbase + lo, Mtot - 1);
    float xa = x[mA], ya = y[mA];

    // y values for the 8 D-matrix rows this lane holds (M = r + 8*hi)
    float yv[8];
#pragma unroll
    for (int r = 0; r < 8; ++r)
        yv[r] = y[min(mbase + r + 8 * hi, Mtot - 1)];

    // pre-pass A: D = a_n + x_m*s_n + y_m*g_n (no shift)
    v2f Ap;
    Ap.x = hi ? 1.0f : xa;
    Ap.y = hi ? 0.0f : ya;

    for (int j = 0; j < 3; ++j) {
        const float4* Tj = &lT[j * NMC];

        // ---- pre-pass: per-m upper bound K' from 128-sample strided max ----
        float km[8];
#pragma unroll
        for (int r = 0; r < 8; ++r) km[r] = -1e30f;
#pragma unroll
        for (int q = 0; q < 8; ++q) {
            float4 tv = Tj[q * 128 + lo];
            v2f B;
            B.x = hi ? tv.z : tv.x;   // K0: s | K2: a
            B.y = hi ? 1.0f : tv.y;   // K1: g | K3: 1
            v8f C = {};
            v8f D = __builtin_amdgcn_wmma_f32_16x16x4_f32(false, Ap, false, B,
                                                          (short)0, C, false, false);
#pragma unroll
            for (int r = 0; r < 8; ++r) km[r] = fmaxf(km[r], D[r]);
        }
#pragma unroll
        for (int r = 0; r < 8; ++r) {
            float m_ = km[r];
#pragma unroll
            for (int off = 1; off < 16; off <<= 1)
                m_ = fmaxf(m_, __shfl_xor(m_, off, 32));
            km[r] = m_;
        }
        if (lane == 0) {
#pragma unroll
            for (int r = 0; r < 8; ++r) ldsK[wave][r] = km[r] + KSLACK;
        } else if (lane == 16) {
#pragma unroll
            for (int r = 0; r < 8; ++r) ldsK[wave][r + 8] = km[r] + KSLACK;
        }
        float kshift = ldsK[wave][lo];

        // shifted A: K3 slot carries -K'_m, so WMMA output is pre-shifted
        v2f A;
        A.x = hi ? 1.0f : xa;
        A.y = hi ? -kshift : ya;

        // ---- main pass: sm = sum_n exp(D) * (1 - exp(-2z)) ----
        float sm[8];
#pragma unroll
        for (int r = 0; r < 8; ++r) sm[r] = 0.0f;

        for (int n0 = 0; n0 < NMC; n0 += 16) {
            float4 tv = Tj[n0 + lo];
            v2f B;
            B.x = hi ? tv.z : tv.x;
            B.y = hi ? 1.0f : tv.y;
            float g2 = tv.w;          // -2*g_n, broadcast for this N column
            v8f C = {};
            v8f D = __builtin_amdgcn_wmma_f32_16x16x4_f32(false, A, false, B,
                                                          (short)0, C, false, false);
#pragma unroll
            for (int r = 0; r < 8; ++r) {
                float t  = __expf(g2 * yv[r]);                 // e^{-2z} in (0,1]
                float e1 = __expf(fminf(D[r], ECLAMP));        // overflow-safe
                sm[r] = fmaf(-e1, t, sm[r] + e1);              // += e1*(1-t)
            }
        }

        // ---- reduce the 16 n-residues across each 16-lane half ----
#pragma unroll
        for (int r = 0; r < 8; ++r) {
            float s_ = sm[r];
#pragma unroll
            for (int off = 1; off < 16; off <<= 1)
                s_ += __shfl_xor(s_, off, 32);
            if (lo == 0)
                lLse[wave][j * 16 + r + 8 * hi] = __logf(s_) + ldsK[wave][r + 8 * hi];
        }
    }
    __syncthreads();

    // ---- combine: lanes 0-15 of each wave each own one m of the tile ----
    float sn = pSn[0];
    float inv_s2 = 1.0f / (sn * sn);
    float local = 0.0f;
    int mC = mbase + lo;
    if (hi == 0 && mC < Mtot) {
        float xm = x[mC], ym = y[mC];
        float ly = __logf(ym), lsn = __logf(sn);
        // b_m (excluded from the WMMA path) re-attached here
        float bm = ly - ym * ym * inv_s2 - 0.5f * xm * xm * inv_s2;
        float base = LOG2F_C + 2.0f * ly + NEG_LGAM15 - 4.0f * lsn
                   - ym * ym * inv_s2 - HALF_L2PI;
        float I1 = pI1[0], I2 = pI2[0], I3 = pI3[0];
        float lp[6];
        lp[0] = base - 0.5f * (xm - I1) * (xm - I1) * inv_s2;
        lp[1] = base - 0.5f * (xm - I2) * (xm - I2) * inv_s2;
        lp[2] = base - 0.5f * (xm - I3) * (xm - I3) * inv_s2;
        lp[3] = bm + lLse[wave][0 * 16 + lo];
        lp[4] = bm + lLse[wave][1 * 16 + lo];
        lp[5] = bm + lLse[wave][2 * 16 + lo];

        float wv[6], wmax = -1e30f;
#pragma unroll
        for (int i = 0; i < 6; ++i) { wv[i] = w[i]; wmax = fmaxf(wmax, wv[i]); }
        float wsum = 0.0f;
#pragma unroll
        for (int i = 0; i < 6; ++i) wsum += __expf(wv[i] - wmax);
        float lw = wmax + __logf(wsum);  // logsumexp(w)

        float tmax = -1e30f, t[6];
#pragma unroll
        for (int i = 0; i < 6; ++i) { t[i] = wv[i] - lw + lp[i]; tmax = fmaxf(tmax, t[i]); }
        float ts = 0.0f;
#pragma unroll
        for (int i = 0; i < 6; ++i) ts += __expf(t[i] - tmax);
        local = -(tmax + __logf(ts));
    }

    red[tid] = local;
    __syncthreads();
    for (int s_ = 128; s_ > 0; s_ >>= 1) {
        if (tid < s_) red[tid] += red[tid + s_];
        __syncthreads();
    }
    if (tid == 0) partial[blockIdx.x] = red[0];
}

// ---------------- kernel 3: deterministic final reduction ----------------
__global__ void arc_reduce(const float* __restrict__ partial, int nb,
                           float* __restrict__ out) {
    __shared__ double red[256];
    double s = 0.0;
    for (int i = threadIdx.x; i < nb; i += 256) s += (double)partial[i];
    red[threadIdx.x] = s;
    __syncthreads();
    for (int k = 128; k > 0; k >>= 1) {
        if (threadIdx.x < k) red[threadIdx.x] += red[threadIdx.x + k];
        __syncthreads();
    }
    if (threadIdx.x == 0) out[0] = (float)red[0];
}

extern "C" void kernel_launch(void* const* d_in, const int* in_sizes, int n_in,
                              void* d_out, int out_size, void* d_ws, size_t ws_size,
                              hipStream_t stream) {
    const float* x    = (const float*)d_in[0];
    const float* y    = (const float*)d_in[1];
    const float* ku12 = (const float*)d_in[2];
    const float* ku23 = (const float*)d_in[3];
    const float* ku13 = (const float*)d_in[4];
    const float* sb   = (const float*)d_in[5];
    const float* snp  = (const float*)d_in[6];
    const float* I1   = (const float*)d_in[7];
    const float* I2   = (const float*)d_in[8];
    const float* I3   = (const float*)d_in[9];
    const float* w    = (const float*)d_in[10];
    int Mtot = in_sizes[0];

    float4* tab4   = (float4*)d_ws;
    float*  partial = (float*)d_ws + 4 * 3 * NMC;

    arc_precompute<<<(3 * NMC + 255) / 256, 256, 0, stream>>>(
        ku12, ku23, ku13, sb, snp, I1, I2, I3, tab4);

    int tiles  = (Mtot + 15) / 16;
    int blocks = (tiles + WPB - 1) / WPB;
    arc_main<<<blocks, 256, 0, stream>>>(x, y, tab4, snp, I1, I2, I3, w,
                                         partial, Mtot);

    arc_reduce<<<1, 256, 0, stream>>>(partial, blocks, (float*)d_out);
}